// GCNEncoder_65704409694423
// MI455X (gfx1250) — compile-verified
//
#include <hip/hip_runtime.h>
#include <hip/hip_bf16.h>

typedef __attribute__((ext_vector_type(2))) float v2f;
typedef __attribute__((ext_vector_type(8))) float v8f;

#define N_NODES 100000
#define N_EDGES 1600000
#define BP_STRIDE 160   // floats per row-pair in swizzled LDS B (padded: no bank aliasing)

// ---------------------------------------------------------------------------
// Normalization: deg (with self loop) -> dis = rsqrt(deg)
// ---------------------------------------------------------------------------
__global__ void k_init_deg(float* __restrict__ deg) {
    int i = blockIdx.x * blockDim.x + threadIdx.x;
    if (i < N_NODES) deg[i] = 1.0f;  // self-loop contribution
}

__global__ void k_deg_accum(const long long* __restrict__ ei, float* __restrict__ deg) {
    int e = blockIdx.x * blockDim.x + threadIdx.x;
    if (e < N_EDGES) {
        int c = (int)ei[(long)N_EDGES + e];  // col = target
        atomicAdd(&deg[c], 1.0f);
    }
}

__global__ void k_rsqrt_inplace(float* __restrict__ d) {
    int i = blockIdx.x * blockDim.x + threadIdx.x;
    if (i < N_NODES) d[i] = rsqrtf(d[i]);   // deg >= 1 always (self loops)
}

// Fuse Wmu [64,32] and Wlv [64,32] into Wcat [64,64] so both latent heads
// share one GEMM and one 64-channel aggregation pass.
__global__ void k_build_wcat(const float* __restrict__ Wmu, const float* __restrict__ Wlv,
                             float* __restrict__ Wcat) {
    int t = blockIdx.x * blockDim.x + threadIdx.x;
    if (t < 64 * 64) {
        int k = t >> 6, c = t & 63;
        Wcat[t] = (c < 32) ? Wmu[k * 32 + c] : Wlv[k * 32 + (c - 32)];
    }
}

// ---------------------------------------------------------------------------
// WMMA fp32 GEMM: C[M,64] = A[M,K] @ B[K,64], K in {128, 64}, M % 16 == 0.
// 256 threads = 8 waves; each wave owns a 16-row tile and all 4 16-col tiles.
// B is staged in LDS in (row-pair, column) swizzled order so each lane's
// fragment {B[r][n], B[r+1][n]} is one aligned ds_load_b64 straight into the
// even-aligned VGPR pair V_WMMA_F32_16X16X4_F32 consumes (no shuffle movs).
// ---------------------------------------------------------------------------
__global__ __launch_bounds__(256) void k_gemm_wmma(const float* __restrict__ A,
                                                   const float* __restrict__ Bm,
                                                   float* __restrict__ C,
                                                   int M, int K) {
    __shared__ __align__(16) float Bs[64 * BP_STRIDE];  // up to 40 KB (K=128)
    const int tid = threadIdx.x;
    const int npair = (K >> 1) * 64;                    // (row-pair, col) elements
    for (int i = tid; i < npair; i += 256) {
        int p = i >> 6;      // row-pair index (rows 2p, 2p+1)
        int n = i & 63;      // column
        float2 v;
        v.x = Bm[(2 * p    ) * 64 + n];
        v.y = Bm[(2 * p + 1) * 64 + n];
        *(float2*)(Bs + p * BP_STRIDE + n * 2) = v;
    }
    __syncthreads();

    const int wave = tid >> 5;
    const int lane = tid & 31;
    const int lh   = lane & 15;          // M (for A) / N (for B,C) within tile
    const int kk   = (lane >> 4) << 1;   // K sub-offset: 0 or 2

    const long mbase = ((long)blockIdx.x * 8 + wave) * 16;
    if (mbase >= M) return;              // wave-uniform guard (EXEC all-1s for WMMA)

    const float* arow  = A + (mbase + lh) * (long)K;
    const float* bbase = Bs + (kk >> 1) * BP_STRIDE + (lh << 1);

    v8f acc0 = {}, acc1 = {}, acc2 = {}, acc3 = {};
    for (int k = 0; k < K; k += 4) {
        // A 16x4 fp32 fragment: lane holds row M=lh, K = k+kk .. k+kk+1
        v2f a = *(const v2f*)(arow + k + kk);
        const float* bp = bbase + (k >> 1) * BP_STRIDE;
        v2f b;
        b = *(const v2f*)(bp);                     // n-tile 0 (cols lh)
        acc0 = __builtin_amdgcn_wmma_f32_16x16x4_f32(false, a, false, b, (short)0, acc0, false, false);
        b = *(const v2f*)(bp + 32);                // n-tile 1 (cols 16+lh)
        acc1 = __builtin_amdgcn_wmma_f32_16x16x4_f32(false, a, false, b, (short)0, acc1, false, false);
        b = *(const v2f*)(bp + 64);                // n-tile 2
        acc2 = __builtin_amdgcn_wmma_f32_16x16x4_f32(false, a, false, b, (short)0, acc2, false, false);
        b = *(const v2f*)(bp + 96);                // n-tile 3
        acc3 = __builtin_amdgcn_wmma_f32_16x16x4_f32(false, a, false, b, (short)0, acc3, false, false);
    }

    // C/D layout: VGPR i -> M = i + (lane>=16 ? 8 : 0), N = lh (+ 16*ntile)
    const long mrow = mbase + (long)((lane >> 4) << 3);
    #pragma unroll
    for (int i = 0; i < 8; ++i) {
        float* crow = C + (mrow + i) * 64 + lh;
        crow[0]  = acc0[i];
        crow[16] = acc1[i];
        crow[32] = acc2[i];
        crow[48] = acc3[i];
    }
}

// ---------------------------------------------------------------------------
// Aggregation: dst starts with self-loop term, then edge-wise atomic scatter.
// Feature maps (25.6 MB) are L2-resident on MI455X (192 MB L2), so the random
// gather + float atomics stay on-chip.
// ---------------------------------------------------------------------------
__global__ void k_selfloop_init(const float* __restrict__ src, const float* __restrict__ dis,
                                float* __restrict__ dst) {
    long t = (long)blockIdx.x * blockDim.x + threadIdx.x;
    if (t < (long)N_NODES * 64) {
        float d = dis[t >> 6];
        dst[t] = src[t] * d * d;   // self-loop norm = dis[i]*dis[i]
    }
}

__global__ void k_edge_scatter(const long long* __restrict__ ei, const float* __restrict__ dis,
                               const float* __restrict__ src, float* __restrict__ dst) {
    long t = (long)blockIdx.x * blockDim.x + threadIdx.x;
    if (t >= (long)N_EDGES * 64) return;
    long e = t >> 6;
    int  c = (int)(t & 63);
    int  r   = (int)ei[e];                      // source (gather row)
    int  col = (int)ei[(long)N_EDGES + e];      // target (scatter row)
    float w = dis[r] * dis[col];
    atomicAdd(dst + (long)col * 64 + c, w * src[(long)r * 64 + c]);
}

__global__ void k_bias_relu(const float* __restrict__ agg, const float* __restrict__ b,
                            float* __restrict__ out) {
    long t = (long)blockIdx.x * blockDim.x + threadIdx.x;
    if (t < (long)N_NODES * 64) {
        float v = agg[t] + b[(int)(t & 63)];
        out[t] = v > 0.0f ? v : 0.0f;
    }
}

// Split fused 64-channel aggregation into mu (cols 0..31) and logvar (32..63),
// add biases, write into concatenated output layout.
__global__ void k_finalize(const float* __restrict__ agg, const float* __restrict__ bmu,
                           const float* __restrict__ blv, float* __restrict__ out) {
    long t = (long)blockIdx.x * blockDim.x + threadIdx.x;
    if (t >= (long)N_NODES * 64) return;
    long i = t >> 6;
    int  c = (int)(t & 63);
    if (c < 32)
        out[i * 32 + c] = agg[t] + bmu[c];
    else
        out[(long)N_NODES * 32 + i * 32 + (c - 32)] = agg[t] + blv[c - 32];
}

// ---------------------------------------------------------------------------
extern "C" void kernel_launch(void* const* d_in, const int* in_sizes, int n_in,
                              void* d_out, int out_size, void* d_ws, size_t ws_size,
                              hipStream_t stream) {
    const float*     x   = (const float*)d_in[0];
    const long long* ei  = (const long long*)d_in[1];   // [2, E] int64
    const float*     W1  = (const float*)d_in[2];
    const float*     b1  = (const float*)d_in[3];
    const float*     Wmu = (const float*)d_in[4];
    const float*     bmu = (const float*)d_in[5];
    const float*     Wlv = (const float*)d_in[6];
    const float*     blv = (const float*)d_in[7];
    float* out = (float*)d_out;

    // Workspace: bufA [N,64], bufB [N,64], dis [N], wcat [64,64]  (~51.6 MB)
    float* ws   = (float*)d_ws;
    float* bufA = ws;
    float* bufB = ws + (size_t)N_NODES * 64;
    float* dis  = ws + (size_t)N_NODES * 128;
    float* wcat = dis + N_NODES;

    const dim3 blk(256);
    const int gN  = (N_NODES + 255) / 256;
    const int gE  = (N_EDGES + 255) / 256;
    const int gNC = (int)(((long)N_NODES * 64 + 255) / 256);
    const int gEC = (int)(((long)N_EDGES * 64 + 255) / 256);
    const int gG  = ((N_NODES / 16) + 7) / 8;   // 6250 tiles / 8 waves per block

    // Normalization
    k_init_deg     <<<gN, blk, 0, stream>>>(dis);
    k_deg_accum    <<<gE, blk, 0, stream>>>(ei, dis);
    k_rsqrt_inplace<<<gN, blk, 0, stream>>>(dis);
    k_build_wcat   <<<16, blk, 0, stream>>>(Wmu, Wlv, wcat);

    // Layer 1: h = relu(agg(x @ W1) + b1)
    k_gemm_wmma    <<<gG,  blk, 0, stream>>>(x, W1, bufA, N_NODES, 128);   // h0 -> A
    k_selfloop_init<<<gNC, blk, 0, stream>>>(bufA, dis, bufB);
    k_edge_scatter <<<gEC, blk, 0, stream>>>(ei, dis, bufA, bufB);
    k_bias_relu    <<<gNC, blk, 0, stream>>>(bufB, b1, bufA);              // h -> A

    // Layer 2+3 fused: [mu|logvar] = agg(h @ [Wmu|Wlv]) + [bmu|blv]
    k_gemm_wmma    <<<gG,  blk, 0, stream>>>(bufA, wcat, bufB, N_NODES, 64); // hcat -> B
    k_selfloop_init<<<gNC, blk, 0, stream>>>(bufB, dis, bufA);
    k_edge_scatter <<<gEC, blk, 0, stream>>>(ei, dis, bufB, bufA);
    k_finalize     <<<gNC, blk, 0, stream>>>(bufA, bmu, blv, out);
}